// RGCN_60026462929254
// MI455X (gfx1250) — compile-verified
//
#include <hip/hip_runtime.h>
#include <hip/hip_bf16.h>

typedef __attribute__((ext_vector_type(2))) float v2f;
typedef __attribute__((ext_vector_type(8))) float v8f;

#define NODE_TILE 128  // rows per block (8 waves x 16 rows)

// ---------------------------------------------------------------------------
// Dense transform:  C[M x NT*16] = A[M x KT*4] @ B[KT*4 x ldb]  (+ bias)
// B is staged in LDS zero-padded to NT*16 columns.  Each wave computes a
// 16-row tile using V_WMMA_F32_16X16X4_F32 (fp32 in, fp32 accumulate).
//
// WMMA f32 16x16x4 layouts (CDNA5 ISA §7.12.2):
//   A (16x4): lane l<16 -> row M=l, VGPR0=K0, VGPR1=K1 ; lanes 16-31 -> K2,K3
//   B (4x16): mirrored: lane l -> col N=l%16, VGPR0=K(hi?2:0), VGPR1=K(hi?3:1)
//   C/D (16x16): VGPR v -> row M = v + (lane>=16 ? 8:0), col N = lane%16
// ---------------------------------------------------------------------------
template <int KT, int NT>
__global__ __launch_bounds__(256) void wmma_transform(
    const float* __restrict__ A,     // [M][KT*4] row-major
    const float* __restrict__ B,     // [KT*4][ldb] row-major
    const float* __restrict__ bias,  // [nbias] (read only if use_bias)
    int use_bias, int nbias,
    float* __restrict__ C,           // [M][NT*16]
    int M, int ldb) {
  constexpr int K = KT * 4;
  constexpr int N = NT * 16;
  __shared__ float sB[K][N];

  // Cooperative load of weights into LDS, zero-padding columns >= ldb.
  for (int i = threadIdx.x; i < K * N; i += 256) {
    int k = i / N, n = i % N;
    sB[k][n] = (n < ldb) ? B[k * ldb + n] : 0.0f;
  }
  __syncthreads();

  const int wave = threadIdx.x >> 5;  // 0..7
  const int lane = threadIdx.x & 31;
  const bool hi = lane >= 16;
  const int row0 = blockIdx.x * NODE_TILE + wave * 16;
  const int mrow = row0 + (lane & 15);
  const int mload = (mrow < M) ? mrow : (M - 1);  // clamp; stores are guarded

  const float* arow = A + (size_t)mload * K;
  __builtin_prefetch(arow, 0, 3);  // global_prefetch_b8

  // Preload all A fragments for this wave's 16-row tile (reused for every nt).
  v2f afrag[KT];
#pragma unroll
  for (int kt = 0; kt < KT; ++kt) {
    const float* ap = arow + kt * 4 + (hi ? 2 : 0);
    afrag[kt].x = ap[0];
    afrag[kt].y = ap[1];
  }

#pragma unroll
  for (int nt = 0; nt < NT; ++nt) {
    v8f acc = {};
#pragma unroll
    for (int kt = 0; kt < KT; ++kt) {
      const int kbase = kt * 4 + (hi ? 2 : 0);
      v2f bfrag;
      bfrag.x = sB[kbase + 0][nt * 16 + (lane & 15)];
      bfrag.y = sB[kbase + 1][nt * 16 + (lane & 15)];
      acc = __builtin_amdgcn_wmma_f32_16x16x4_f32(
          /*neg_a=*/false, afrag[kt], /*neg_b=*/false, bfrag,
          /*c_mod=*/(short)0, acc, /*reuse_a=*/false, /*reuse_b=*/false);
    }
    const int col = nt * 16 + (lane & 15);
    float bv = 0.0f;
    if (use_bias && col < nbias) bv = bias[col];
#pragma unroll
    for (int v = 0; v < 8; ++v) {
      const int r = row0 + v + (hi ? 8 : 0);
      if (r < M) C[(size_t)r * N + col] = acc[v] + bv;
    }
  }
}

// ---------------------------------------------------------------------------
// Per-(relation,dst) edge counts: cnt[r*N + dst] += 1.0   (shared by layers)
// ---------------------------------------------------------------------------
__global__ __launch_bounds__(256) void count_edges(
    const int* __restrict__ ei, const int* __restrict__ et,
    float* __restrict__ cnt, int E, int N) {
  int e = blockIdx.x * blockDim.x + threadIdx.x;
  if (e >= E) return;
  int dst = ei[E + e];  // edge_index[1][e]
  int r = et[e];
  atomicAdd(&cnt[(size_t)r * N + dst], 1.0f);
}

// ---------------------------------------------------------------------------
// Gather h[type][src] and scatter-add into agg[type][dst].
// C/4 threads per edge, float4 per thread -> coalesced 4*C bytes per edge.
// ---------------------------------------------------------------------------
template <int C>
__global__ __launch_bounds__(256) void scatter_edges(
    const int* __restrict__ ei, const int* __restrict__ et,
    const float* __restrict__ h,  // [2][N][C]
    float* __restrict__ agg,      // [2][N][C]
    int E, int N) {
  constexpr int G = C / 4;
  long long t = (long long)blockIdx.x * blockDim.x + threadIdx.x;
  if (t >= (long long)E * G) return;
  int e = (int)(t / G);
  int g = (int)(t % G);
  int src = ei[e];
  int dst = ei[E + e];
  int r = et[e];
  const float4 v =
      *(const float4*)(h + ((size_t)r * N + src) * C + (size_t)g * 4);
  float* ap = agg + ((size_t)r * N + dst) * C + (size_t)g * 4;
  atomicAdd(ap + 0, v.x);
  atomicAdd(ap + 1, v.y);
  atomicAdd(ap + 2, v.z);
  atomicAdd(ap + 3, v.w);
}

// ---------------------------------------------------------------------------
// out[n][c<outC] = (relu?) self[n][c] + agg0[n][c]/max(cnt0,1) + agg1/...
// ---------------------------------------------------------------------------
template <int C, bool RELU>
__global__ __launch_bounds__(256) void combine_nodes(
    const float* __restrict__ hself,  // [N][C]
    const float* __restrict__ agg,    // [2][N][C]
    const float* __restrict__ cnt,    // [2][N]
    float* __restrict__ out,          // [N][outC]
    int N, int outC) {
  long long t = (long long)blockIdx.x * blockDim.x + threadIdx.x;
  if (t >= (long long)N * C) return;
  int n = (int)(t / C);
  int c = (int)(t % C);
  float inv0 = 1.0f / fmaxf(cnt[n], 1.0f);
  float inv1 = 1.0f / fmaxf(cnt[(size_t)N + n], 1.0f);
  float v = hself[t] + agg[t] * inv0 + agg[(size_t)N * C + t] * inv1;
  if (RELU) v = fmaxf(v, 0.0f);
  if (c < outC) out[(size_t)n * outC + c] = v;
}

extern "C" void kernel_launch(void* const* d_in, const int* in_sizes, int n_in,
                              void* d_out, int out_size, void* d_ws,
                              size_t ws_size, hipStream_t stream) {
  const float* x     = (const float*)d_in[0];  // [N][128]
  const int*   ei    = (const int*)d_in[1];    // [2][E]
  const int*   et    = (const int*)d_in[2];    // [E]
  const float* W1    = (const float*)d_in[3];  // [2][128][64]
  const float* root1 = (const float*)d_in[4];  // [128][64]
  const float* b1    = (const float*)d_in[5];  // [64]
  const float* W2    = (const float*)d_in[6];  // [2][64][10]
  const float* root2 = (const float*)d_in[7];  // [64][10]
  const float* b2    = (const float*)d_in[8];  // [10]
  float* out = (float*)d_out;

  const int N = in_sizes[0] / 128;  // 50000 nodes
  const int E = in_sizes[2];        // 800000 edges

  // Workspace layout (floats).
  float* ws = (float*)d_ws;
  float* H1self = ws; ws += (size_t)N * 64;      // x@root1 + b1
  float* H1r    = ws; ws += (size_t)2 * N * 64;  // x@W1[r]
  float* agg1   = ws; ws += (size_t)2 * N * 64;
  float* cnt    = ws; ws += (size_t)2 * N;
  float* hrelu  = ws; ws += (size_t)N * 64;
  float* H2self = ws; ws += (size_t)N * 16;      // padded N=16
  float* H2r    = ws; ws += (size_t)2 * N * 16;
  float* agg2   = ws; ws += (size_t)2 * N * 16;

  // Zero the accumulation buffers (graph-capturable memset nodes).
  hipMemsetAsync(agg1, 0, (size_t)2 * N * 64 * sizeof(float), stream);
  hipMemsetAsync(cnt, 0, (size_t)2 * N * sizeof(float), stream);
  hipMemsetAsync(agg2, 0, (size_t)2 * N * 16 * sizeof(float), stream);

  const int gblk = (N + NODE_TILE - 1) / NODE_TILE;

  // ---- Layer 1 dense transforms (K=128 -> KT=32, Ncols=64 -> NT=4) ----
  wmma_transform<32, 4><<<gblk, 256, 0, stream>>>(x, root1, b1, 1, 64, H1self,
                                                  N, 64);
  wmma_transform<32, 4><<<gblk, 256, 0, stream>>>(x, W1, b1, 0, 0, H1r, N, 64);
  wmma_transform<32, 4><<<gblk, 256, 0, stream>>>(
      x, W1 + (size_t)128 * 64, b1, 0, 0, H1r + (size_t)N * 64, N, 64);

  // ---- Edge counts (shared by both layers) ----
  count_edges<<<(E + 255) / 256, 256, 0, stream>>>(ei, et, cnt, E, N);

  // ---- Layer 1 message aggregation ----
  {
    long long T = (long long)E * 16;  // C=64 -> 16 threads/edge
    scatter_edges<64><<<(unsigned)((T + 255) / 256), 256, 0, stream>>>(
        ei, et, H1r, agg1, E, N);
  }
  {
    long long T = (long long)N * 64;
    combine_nodes<64, true><<<(unsigned)((T + 255) / 256), 256, 0, stream>>>(
        H1self, agg1, cnt, hrelu, N, 64);
  }

  // ---- Layer 2 dense transforms (K=64 -> KT=16, Ncols=10 padded to 16) ----
  wmma_transform<16, 1><<<gblk, 256, 0, stream>>>(hrelu, root2, b2, 1, 10,
                                                  H2self, N, 10);
  wmma_transform<16, 1><<<gblk, 256, 0, stream>>>(hrelu, W2, b2, 0, 0, H2r, N,
                                                  10);
  wmma_transform<16, 1><<<gblk, 256, 0, stream>>>(
      hrelu, W2 + (size_t)64 * 10, b2, 0, 0, H2r + (size_t)N * 16, N, 10);

  // ---- Layer 2 message aggregation ----
  {
    long long T = (long long)E * 4;  // C=16 -> 4 threads/edge
    scatter_edges<16><<<(unsigned)((T + 255) / 256), 256, 0, stream>>>(
        ei, et, H2r, agg2, E, N);
  }
  {
    long long T = (long long)N * 16;
    combine_nodes<16, false><<<(unsigned)((T + 255) / 256), 256, 0, stream>>>(
        H2self, agg2, cnt, out, N, 10);
  }
}